// PointWarping_57767310131684
// MI455X (gfx1250) — compile-verified
//
#include <hip/hip_runtime.h>
#include <math.h>

#define N_PTS 8192   // N1 == N2
#define B_SZ  2
#define SPLITS 4     // waves cooperating per 16-query tile
#define SRC_PER_SPLIT (N_PTS / SPLITS)   // 2048
#define QT_PER_BLOCK 2                   // 2 query tiles * 4 splits = 8 waves/block

typedef __attribute__((ext_vector_type(2))) float v2f;
typedef __attribute__((ext_vector_type(8))) float v8f;

// Branchless sorted insert into a 3-element min-list (d0 <= d1 <= d2).
__device__ __forceinline__ void insert3(float d, int idx,
                                        float& d0, float& d1, float& d2,
                                        int& i0, int& i1, int& i2) {
  bool c0 = d < d0, c1 = d < d1, c2 = d < d2;
  float nd1 = c0 ? d0 : (c1 ? d : d1);
  int   ni1 = c0 ? i0 : (c1 ? idx : i1);
  float nd2 = c1 ? d1 : (c2 ? d : d2);
  int   ni2 = c1 ? i1 : (c2 ? idx : i2);
  d0 = c0 ? d : d0;  i0 = c0 ? idx : i0;
  d1 = nd1;          i1 = ni1;
  d2 = nd2;          i2 = ni2;
}

// Precompute augmented source rows: s = xyz1 + flow1 (warped source),
// aug = (-2sx, -2sy, -2sz, |s|^2)  -> one b128 load per 16x16 tile lane.
__global__ __launch_bounds__(256) void pw_prep(const float* __restrict__ xyz1,
                                               const float* __restrict__ flow1,
                                               float4* __restrict__ aug) {
  int i = blockIdx.x * blockDim.x + threadIdx.x;   // [0, B*N)
  int b = i >> 13;
  int j = i & (N_PTS - 1);
  const float* X = xyz1 + b * 3 * N_PTS;
  const float* F = flow1 + b * 3 * N_PTS;
  float sx = X[j] + F[j];
  float sy = X[N_PTS + j] + F[N_PTS + j];
  float sz = X[2 * N_PTS + j] + F[2 * N_PTS + j];
  aug[i] = make_float4(-2.0f * sx, -2.0f * sy, -2.0f * sz,
                       sx * sx + sy * sy + sz * sz);
}

// Process one 16-source x 16-query WMMA tile and fold into the running top-3.
__device__ __forceinline__ void tile_step(float4 a4, int t, int hl, v2f bm,
                                          float& d0, float& d1, float& d2,
                                          int& i0, int& i1, int& i2) {
  v2f am;
  am.x = hl ? a4.z : a4.x;               // K=2 : K=0
  am.y = hl ? a4.w : a4.y;               // K=3 : K=1
  v8f c = {};
  c = __builtin_amdgcn_wmma_f32_16x16x4_f32(false, am, false, bm,
                                            (short)0, c, false, false);
  // This lane holds scores of its query vs sources m = hl*8 + r of tile t.
  float m0 = fminf(fminf(c[0], c[1]), fminf(c[2], c[3]));
  float m1 = fminf(fminf(c[4], c[5]), fminf(c[6], c[7]));
  if (fminf(m0, m1) < d2) {              // quick reject: v_cmpx + execz skip
    const int base = t + hl * 8;
#pragma unroll
    for (int r = 0; r < 8; ++r)
      insert3(c[r], base + r, d0, d1, d2, i0, i1, i2);
  }
}

// 4 waves per 16-query tile, each scanning 2048 sources (split-K over sources),
// top-3 lists merged through LDS. Loads double-buffered: 2 b128 in flight.
__global__ __launch_bounds__(256) void pw_knn_warp(const float* __restrict__ xyz2,
                                                   const float* __restrict__ flow1,
                                                   const float4* __restrict__ aug,
                                                   float* __restrict__ out) {
  __shared__ float sd[QT_PER_BLOCK][SPLITS][3][16];
  __shared__ int   si[QT_PER_BLOCK][SPLITS][3][16];

  const int wib   = threadIdx.x >> 5;    // wave in block: 0..7
  const int lane  = threadIdx.x & 31;
  const int qt    = wib >> 2;            // local query tile: 0..1
  const int split = wib & 3;             // source-range split: 0..3
  const int g     = blockIdx.x * QT_PER_BLOCK + qt;   // global tile: 0..1023
  const int b     = g >> 9;              // batch
  const int qtile = g & 511;
  const int col = lane & 15;             // query within tile
  const int hl  = lane >> 4;             // half-wave: K{0,1}/rows 0-7 vs K{2,3}/rows 8-15
  const int q   = qtile * 16 + col;

  const float* X2 = xyz2 + b * 3 * N_PTS;
  const float qx = X2[q];
  const float qy = X2[N_PTS + q];
  const float qz = X2[2 * N_PTS + q];

  // B matrix (queries), fixed for the whole loop: col n = [qx,qy,qz,1]
  v2f bm;
  bm.x = hl ? qz   : qx;
  bm.y = hl ? 1.0f : qy;

  float d0 = 3.402823466e+38f, d1 = d0, d2 = d0;
  int i0 = 0, i1 = 0, i2 = 0;

  const float4* A = aug + b * N_PTS;
  const int t0 = split * SRC_PER_SPLIT;
  const int t1 = t0 + SRC_PER_SPLIT;

  // 2 tiles (32 sources) per iteration; both loads issued before first WMMA.
  for (int t = t0; t < t1; t += 32) {
    float4 a4a = A[t + col];
    float4 a4b = A[t + 16 + col];
    tile_step(a4a, t,      hl, bm, d0, d1, d2, i0, i1, i2);
    tile_step(a4b, t + 16, hl, bm, d0, d1, d2, i0, i1, i2);
  }

  // Merge the two half-wave lists (sources split low/high rows per tile).
  float e0 = __shfl_xor(d0, 16, 32);
  float e1 = __shfl_xor(d1, 16, 32);
  float e2 = __shfl_xor(d2, 16, 32);
  int   j0 = __shfl_xor(i0, 16, 32);
  int   j1 = __shfl_xor(i1, 16, 32);
  int   j2 = __shfl_xor(i2, 16, 32);
  insert3(e0, j0, d0, d1, d2, i0, i1, i2);
  insert3(e1, j1, d0, d1, d2, i0, i1, i2);
  insert3(e2, j2, d0, d1, d2, i0, i1, i2);

  // Publish per-split top-3 to LDS.
  if (hl == 0) {
    sd[qt][split][0][col] = d0;  si[qt][split][0][col] = i0;
    sd[qt][split][1][col] = d1;  si[qt][split][1][col] = i1;
    sd[qt][split][2][col] = d2;  si[qt][split][2][col] = i2;
  }
  __syncthreads();

  // Split 0 merges all 4 lists (ascending split => ascending indices, so
  // strict-< keeps first occurrence like top_k) and writes the output.
  if (split == 0 && hl == 0) {
#pragma unroll
    for (int s = 1; s < SPLITS; ++s)
#pragma unroll
      for (int k = 0; k < 3; ++k)
        insert3(sd[qt][s][k][col], si[qt][s][k][col], d0, d1, d2, i0, i1, i2);

    // Gather 3 neighbors, recompute exact distances (reference numerics),
    // inverse-distance weights, blend flows, write q - flow2 ([B,3,N]).
    const float* F = flow1 + b * 3 * N_PTS;
    int idx[3] = { i0, i1, i2 };
    float wsum = 0.0f, fx = 0.0f, fy = 0.0f, fz = 0.0f;
#pragma unroll
    for (int k = 0; k < 3; ++k) {
      float4 a4 = A[idx[k]];
      float sx = -0.5f * a4.x, sy = -0.5f * a4.y, sz = -0.5f * a4.z;
      float dx = sx - qx, dy = sy - qy, dz = sz - qz;
      float dist = sqrtf(dx * dx + dy * dy + dz * dz);
      dist = fmaxf(dist, 1e-10f);
      float w = 1.0f / dist;
      wsum += w;
      fx += w * F[idx[k]];
      fy += w * F[N_PTS + idx[k]];
      fz += w * F[2 * N_PTS + idx[k]];
    }
    float inv = 1.0f / wsum;
    float* O = out + b * 3 * N_PTS;
    O[q]             = qx - fx * inv;
    O[N_PTS + q]     = qy - fy * inv;
    O[2 * N_PTS + q] = qz - fz * inv;
  }
}

extern "C" void kernel_launch(void* const* d_in, const int* in_sizes, int n_in,
                              void* d_out, int out_size, void* d_ws, size_t ws_size,
                              hipStream_t stream) {
  (void)in_sizes; (void)n_in; (void)out_size; (void)ws_size;
  const float* xyz1  = (const float*)d_in[0];   // [B,3,N1]
  const float* xyz2  = (const float*)d_in[1];   // [B,3,N2]
  const float* flow1 = (const float*)d_in[2];   // [B,3,N1]
  float* out = (float*)d_out;                   // [B,3,N2]
  float4* aug = (float4*)d_ws;                  // B*N1 float4 = 256 KB scratch

  // Prep: B*N = 16384 sources -> 64 blocks of 256.
  pw_prep<<<(B_SZ * N_PTS) / 256, 256, 0, stream>>>(xyz1, flow1, aug);

  // Main: 1024 query tiles x 4 splits = 4096 waves; 8 waves/block -> 512 blocks.
  pw_knn_warp<<<(B_SZ * (N_PTS / 16) * SPLITS) / 8, 256, 0, stream>>>(
      xyz2, flow1, aug, out);
}